// CtoE_58016418234798
// MI455X (gfx1250) — compile-verified
//
#include <hip/hip_runtime.h>

typedef float v2f __attribute__((ext_vector_type(2)));
typedef float v8f __attribute__((ext_vector_type(8)));

// Problem dimensions (fixed by reference setup_inputs)
constexpr int BS   = 4;
constexpr int NM   = 512;
constexpr int DC   = 64;
constexpr int DE   = 128;
constexpr int MA   = 16;            // M_ATOMS
constexpr int NB   = NM / MA;       // 32 blocks per dim
constexpr int ROWS = BS * NB * NB;  // 4096 pooled "rows"
constexpr int KDIM = 4 * DC;        // 256 concat features
constexpr float EPSV = 1e-8f;
constexpr float BIGV = 100000.0f;

// ---------------------------------------------------------------------------
// Kernel 1: single-pass masked block pooling (mean / min / max / var)
// One 256-thread block per (b, i, j) tile; each tile reads 16*16*64 f32 = 64KB
// with perfectly coalesced float4 loads (one contiguous 4KB row per iter).
// ---------------------------------------------------------------------------
__global__ __launch_bounds__(256) void pool_kernel(
    const float* __restrict__ delta1,
    const float* __restrict__ cm1,
    const float* __restrict__ cm2,
    float* __restrict__ z)
{
    const int bidx = blockIdx.x;             // 0..ROWS-1
    const int b    = bidx / (NB * NB);
    const int ij   = bidx % (NB * NB);
    const int bi   = ij / NB;
    const int bj   = ij % NB;

    const int t  = threadIdx.x;              // 0..255
    const int q  = t & 15;                   // channel quad: c = 4q..4q+3
    const int pg = t >> 4;                   // column-atom index (0..15)

    __shared__ float s_cm1[MA];
    __shared__ float s_cm2[MA];
    if (t < MA)          s_cm1[t]      = cm1[b * NM + bi * MA + t];
    else if (t < 2 * MA) s_cm2[t - MA] = cm2[b * NM + bj * MA + (t - MA)];
    __syncthreads();

    float raw[4] = {0.f, 0.f, 0.f, 0.f};
    float smd[4] = {0.f, 0.f, 0.f, 0.f};
    float sm2[4] = {0.f, 0.f, 0.f, 0.f};
    float vmn[4] = { 3.4e38f,  3.4e38f,  3.4e38f,  3.4e38f};
    float vmx[4] = {-3.4e38f, -3.4e38f, -3.4e38f, -3.4e38f};
    float cnt = 0.f;

    const float cm2v = s_cm2[pg];
    // iterate over the 16 row-atoms; thread owns (row a, col pg, channels 4q..4q+3)
    #pragma unroll 4
    for (int a = 0; a < MA; ++a) {
        const size_t off = ((size_t)(b * NM + bi * MA + a) * NM
                            + (size_t)(bj * MA + pg)) * DC + 4 * q;
        const float4 dv = *reinterpret_cast<const float4*>(delta1 + off);
        const float msk = s_cm1[a] * cm2v;
        const float imk = 1.f - msk;
        const float d[4] = {dv.x, dv.y, dv.z, dv.w};
        cnt += msk;
        #pragma unroll
        for (int k = 0; k < 4; ++k) {
            raw[k] += d[k];
            smd[k] += msk * d[k];
            sm2[k] += msk * d[k] * d[k];
            vmn[k]  = fminf(vmn[k], d[k] + BIGV * imk);
            vmx[k]  = fmaxf(vmx[k], d[k] - BIGV * imk);
        }
    }

    // LDS tree reduction across the 16 pair-groups (same q combine)
    __shared__ float red[5][256][4];   // raw, smd, sm2, min, max
    __shared__ float rcnt[256];
    #pragma unroll
    for (int k = 0; k < 4; ++k) {
        red[0][t][k] = raw[k];
        red[1][t][k] = smd[k];
        red[2][t][k] = sm2[k];
        red[3][t][k] = vmn[k];
        red[4][t][k] = vmx[k];
    }
    rcnt[t] = cnt;
    __syncthreads();

    #pragma unroll
    for (int off = 128; off >= 16; off >>= 1) {
        if (t < off) {
            #pragma unroll
            for (int k = 0; k < 4; ++k) {
                red[0][t][k] += red[0][t + off][k];
                red[1][t][k] += red[1][t + off][k];
                red[2][t][k] += red[2][t + off][k];
                red[3][t][k]  = fminf(red[3][t][k], red[3][t + off][k]);
                red[4][t][k]  = fmaxf(red[4][t][k], red[4][t + off][k]);
            }
            rcnt[t] += rcnt[t + off];
        }
        __syncthreads();
    }

    if (t < 16) {
        const float c   = rcnt[t];
        const float div = c + EPSV;
        float* zr = z + (size_t)bidx * KDIM;
        #pragma unroll
        for (int k = 0; k < 4; ++k) {
            const float mu  = red[0][t][k] / div;
            const float var = (red[2][t][k] - 2.f * mu * red[1][t][k]
                               + mu * mu * c) / div;
            zr[          4 * t + k] = mu;            // mean
            zr[ DC     + 4 * t + k] = red[3][t][k];  // masked min
            zr[2 * DC  + 4 * t + k] = red[4][t][k];  // masked max
            zr[3 * DC  + 4 * t + k] = var;           // masked variance
        }
    }
}

// ---------------------------------------------------------------------------
// Kernel 2: out = Z (4096x256) @ W^T (256x128) + b via V_WMMA_F32_16X16X4_F32
// One wave32 per 16x16 output tile; 64 WMMA steps over K.
// A layout (16x4 f32): v0 = K=(0|2) by lane half, v1 = K=(1|3); B mirrors.
// D layout: VGPR r -> M = r + 8*half, N = lane%16.
// ---------------------------------------------------------------------------
__global__ __launch_bounds__(256) void gemm_kernel(
    const float* __restrict__ z,
    const float* __restrict__ W,     // (DE, KDIM) row-major: W[n][k]
    const float* __restrict__ bias,  // (DE)
    float* __restrict__ out)         // (ROWS, DE)
{
    const int wave = (blockIdx.x << 3) + (threadIdx.x >> 5);  // 0..2047
    const int rt   = wave >> 3;        // row tile  (0..255)
    const int ct   = wave & 7;         // col tile  (0..7)
    const int lane = threadIdx.x & 31;
    const int half = lane >> 4;
    const int lm   = lane & 15;
    const int row0 = rt * 16;
    const int col0 = ct * 16;

    const float* zr = z + (size_t)(row0 + lm) * KDIM;   // A row (M = lm)
    const float* wr = W + (size_t)(col0 + lm) * KDIM;   // B col (N = lm), W[n][k]

    v8f acc = {};
    #pragma unroll 8
    for (int kk = 0; kk < KDIM; kk += 4) {
        const v2f a  = *reinterpret_cast<const v2f*>(zr + kk + 2 * half);
        const v2f bb = *reinterpret_cast<const v2f*>(wr + kk + 2 * half);
        acc = __builtin_amdgcn_wmma_f32_16x16x4_f32(
            /*neg_a=*/false, a, /*neg_b=*/false, bb,
            /*c_mod=*/(short)0, acc, /*reuse_a=*/false, /*reuse_b=*/false);
    }

    const float bv = bias[col0 + lm];
    #pragma unroll
    for (int r = 0; r < 8; ++r) {
        const int row = row0 + (half << 3) + r;
        out[(size_t)row * DE + col0 + lm] = acc[r] + bv;
    }
}

// ---------------------------------------------------------------------------
extern "C" void kernel_launch(void* const* d_in, const int* in_sizes, int n_in,
                              void* d_out, int out_size, void* d_ws, size_t ws_size,
                              hipStream_t stream)
{
    (void)in_sizes; (void)n_in; (void)out_size; (void)ws_size;
    const float* delta1 = (const float*)d_in[0];
    const float* cm1    = (const float*)d_in[1];
    const float* cm2    = (const float*)d_in[2];
    const float* W      = (const float*)d_in[3];
    const float* bias   = (const float*)d_in[4];
    float* out = (float*)d_out;
    float* z   = (float*)d_ws;   // ROWS * KDIM * 4 bytes = 4 MB scratch

    pool_kernel<<<ROWS, 256, 0, stream>>>(delta1, cm1, cm2, z);

    // (ROWS/16)*(DE/16) = 2048 tiles, 8 waves per 256-thread block -> 256 blocks
    gemm_kernel<<<256, 256, 0, stream>>>(z, W, bias, out);
}